// TreeGrammar_51118700757558
// MI455X (gfx1250) — compile-verified
//
#include <hip/hip_runtime.h>
#include <math.h>

// Problem sizes (fixed by the reference)
#define MDIM 1024
#define BDIM 8192

typedef __attribute__((ext_vector_type(2))) float v2f;
typedef __attribute__((ext_vector_type(8))) float v8f;

// ---------------------------------------------------------------------------
// Kernel 1: out[b] = dot(x[b,:], W_base) + (b_base + b_plus + b_prod)
// Uses V_WMMA_F32_16X16X4_F32. One wave handles 16 batch rows; W_base is
// replicated across all 16 N-columns so every column of D carries the same
// GEMV partial (reduction runs on the matrix pipe; pass is HBM-bandwidth
// bound anyway and leaves x hot in the 192MB L2 for the scan kernel).
//
// f32 A-matrix (16x4) layout: lanes 0-15 rows 0-15 K={0,1}, lanes 16-31 K={2,3}
// f32 B-matrix (4x16) layout: VGPR0/1 hold K={0,1} (lanes 0-15), K={2,3} (16-31)
// f32 D (16x16): lanes 0-15 col N=lane rows 0-7 in c[0..7]; lanes 16-31 rows 8-15
// ---------------------------------------------------------------------------
__global__ __launch_bounds__(256) void base_gemv_wmma(
    const float* __restrict__ x, const float* __restrict__ Wb,
    const float* __restrict__ bb, const float* __restrict__ bp,
    const float* __restrict__ br, float* __restrict__ out)
{
    const int lane = threadIdx.x & 31;
    const int warp = threadIdx.x >> 5;
    const int half = lane >> 4;      // which K pair this lane supplies
    const int e    = lane & 15;      // row within tile (A) / column (D)
    const int b0   = (blockIdx.x * 8 + warp) * 16;

    const float* xr = x  + (size_t)(b0 + e) * MDIM + 2 * half;
    const float* wr = Wb + 2 * half;

    v8f c = {0.f, 0.f, 0.f, 0.f, 0.f, 0.f, 0.f, 0.f};
#pragma unroll 8
    for (int k0 = 0; k0 < MDIM; k0 += 4) {
        v2f a = *(const v2f*)(xr + k0);           // x[row][k0+2h], x[row][k0+2h+1]
        v2f b = *(const v2f*)(wr + k0);           // W[k0+2h], W[k0+2h+1] (all N cols)
        c = __builtin_amdgcn_wmma_f32_16x16x4_f32(
                /*neg_a=*/false, a, /*neg_b=*/false, b,
                /*c_mod=*/(short)0, c, /*reuse_a=*/false, /*reuse_b=*/false);
    }

    const float bias = bb[0] + bp[0] + br[0];
    float cr[8];
#pragma unroll
    for (int q = 0; q < 8; ++q) cr[q] = c[q];
    // every D column is identical; lane (half,e<8) holds row half*8+e in cr[e]
    if (e < 8) out[b0 + half * 8 + e] = cr[e] + bias;
}

// ---------------------------------------------------------------------------
// Kernel 2: sequential tree scan, 32 batch elements per 64-thread workgroup.
// Each wave32 owns 16 batch elements: lanes 0-15 compute "plus" (sigmoid),
// lanes 16-31 compute "prod". Carry lives in LDS, private per wave -> no
// barriers in the 1024-step loop (LDS ops are in-order per wave on CDNA5).
// Stride-17 padding keeps the 16-lane carry rows bank-conflict free.
// x-gathers (t = -1) go to global (L2-resident after kernel 1) through a
// branchless pointer select (flat load serves either aperture).
// LDS: 2 waves * 2048*17 f32 carry + 4096 int idx + 5120 f32 coeffs = 308 KB.
// ---------------------------------------------------------------------------
#define SMEM_FLOATS (2 * 2048 * 17 + 2048 + 2048 + 3 * 1024 + 2048)
#define SMEM_BYTES  (SMEM_FLOATS * 4)   // 315392 B <= 320 KB WGP LDS

__global__ __launch_bounds__(64) void tree_scan(
    const float* __restrict__ x, const int* __restrict__ raw,
    const float* __restrict__ Wsig, const float* __restrict__ bsig,
    const float* __restrict__ wplus, const float* __restrict__ wprod,
    float* __restrict__ out)
{
    extern __shared__ float smem[];
    float* carry = smem;                        // [2][2048*17]
    int*   idxA  = (int*)(smem + 2 * 34816);    // [2*1024]  (j*1024+i)
    int*   idxB  = idxA + 2048;                 // [2*1024]
    float* ws0   = (float*)(idxB + 2048);       // [1024]
    float* ws1   = ws0 + 1024;                  // [1024]
    float* bsl   = ws1 + 1024;                  // [1024]
    float* wc    = bsl + 1024;                  // [2*1024]  (j*1024+i)

    const int tid = threadIdx.x;

    // Stage index + coefficient tables once (cooperative, coalesced).
    for (int t = tid; t < 2048; t += 64) {
        const int jj = t >> 10, ii = t & 1023;
        idxA[t] = raw[(jj * 1024 + ii) * 2 + 0];
        idxB[t] = raw[(jj * 1024 + ii) * 2 + 1];
    }
    for (int i = tid; i < 1024; i += 64) {
        ws0[i]       = Wsig[2 * i];
        ws1[i]       = Wsig[2 * i + 1];
        bsl[i]       = bsig[i];
        wc[i]        = wplus[i];
        wc[1024 + i] = wprod[i];
    }
    __syncthreads();

    const int lane = tid & 31;
    const int wid  = tid >> 5;            // wave within workgroup
    const int e    = lane & 15;           // batch element within wave
    const int j    = (lane >> 4) & 1;     // 0 = plus op, 1 = prod op
    const int row  = blockIdx.x * 32 + wid * 16 + e;
    const float* xrow = x + (size_t)row * MDIM;
    float* cw = carry + wid * 34816;      // this wave's private carry region
    const int jo = j * 1024;

    float acc = 0.f;
    for (int i = 0; i < MDIM; ++i) {
        const int rA = idxA[jo + i];
        const int rB = idxB[jo + i];
        // raw code r: r<1024 -> x[:,r]; 1024..2047 -> plus[i]; 2048..3071 -> prod[i]
        const float* pA = (rA < 1024) ? (xrow + rA) : (cw + (rA - 1024) * 17 + e);
        const float* pB = (rB < 1024) ? (xrow + rB) : (cw + (rB - 1024) * 17 + e);
        const float vA = *pA;
        const float vB = *pB;
        const float z  = fmaf(ws0[i], vA, fmaf(ws1[i], vB, bsl[i]));
        const float sg = 1.0f / (1.0f + __expf(-z));
        const float res = (j == 0) ? sg : (vA * vB);
        cw[(jo + i) * 17 + e] = res;          // single ds_store after reconvergence
        acc = fmaf(res, wc[jo + i], acc);     // fold the output dot product in
    }

    // merge plus (lanes 0-15) and prod (lanes 16-31) accumulators
    const float accO = __shfl_xor(acc, 16, 32);
    if (j == 0) out[row] += acc + accO;       // kernel 1 already wrote base+biases
}

// ---------------------------------------------------------------------------
extern "C" void kernel_launch(void* const* d_in, const int* in_sizes, int n_in,
                              void* d_out, int out_size, void* d_ws, size_t ws_size,
                              hipStream_t stream)
{
    const float* x    = (const float*)d_in[0];   // (B, M)
    const int*   raw  = (const int*)  d_in[1];   // (2, M, 2)
    const float* Wb   = (const float*)d_in[2];   // (1, M)
    const float* bb   = (const float*)d_in[3];   // (1,)
    const float* Wsig = (const float*)d_in[4];   // (M, 2)
    const float* bsig = (const float*)d_in[5];   // (M,)
    const float* wp   = (const float*)d_in[6];   // (M,)
    const float* bp   = (const float*)d_in[7];   // (1,)
    const float* wr   = (const float*)d_in[8];   // (M,)
    const float* br   = (const float*)d_in[9];   // (1,)
    float* out = (float*)d_out;                  // (B, 1)

    // Pass 1: WMMA f32 GEMV -> out = x@Wb^T + (b_base+b_plus+b_prod); warms L2 with x.
    base_gemv_wmma<<<BDIM / (16 * 8), 256, 0, stream>>>(x, Wb, bb, bp, br, out);
    // Pass 2: LDS-resident sequential scan + fused output dots.
    tree_scan<<<BDIM / 32, 64, SMEM_BYTES, stream>>>(x, raw, Wsig, bsig, wp, wr, out);
}